// SupervisedBarycentricMapping_3281355014644
// MI455X (gfx1250) — compile-verified
//
#include <hip/hip_runtime.h>
#include <cmath>

typedef __attribute__((ext_vector_type(2))) float v2f;
typedef __attribute__((ext_vector_type(4))) float v4f;
typedef __attribute__((ext_vector_type(8))) float v8f;

#define kN 8192
#define kD 256
#define kDL 16
#define kITER 50
#define NEG_INF (-3.402823466e38f)

static __device__ __forceinline__ v8f wmma_f32(v2f a, v2f b, v8f c) {
  // V_WMMA_F32_16X16X4_F32 : D = A(16x4) x B(4x16) + C(16x16)
  return __builtin_amdgcn_wmma_f32_16x16x4_f32(false, a, false, b, (short)0, c,
                                               false, false);
}

// branchless online logsumexp update: one transcendental per element
static __device__ __forceinline__ void lse_update(float x, float& mx,
                                                  float& s) {
  float d = x - mx;
  float e = __expf(-fabsf(d));
  float s_hi = __fmaf_rn(s, e, 1.0f);  // x is the new max
  float s_lo = s + e;                  // x <= mx
  bool gt = d > 0.0f;
  s = gt ? s_hi : s_lo;
  mx = gt ? x : mx;
}

// merge two (max, sum) pairs
static __device__ __forceinline__ void lse_merge(float m2, float s2, float& mx,
                                                 float& s) {
  float mn = fmaxf(mx, m2);
  s = s * __expf(mx - mn) + s2 * __expf(m2 - mn);
  mx = mn;
}

// ---------------------------------------------------------------- row sum-sq
__global__ __launch_bounds__(256) void row_sumsq(const float* __restrict__ src,
                                                 float* __restrict__ dst,
                                                 int ncols) {
  int lane = threadIdx.x & 31;
  int wave = threadIdx.x >> 5;
  int row = blockIdx.x * 8 + wave;
  const float* p = src + (size_t)row * ncols;
  float s = 0.0f;
  for (int k = lane; k < ncols; k += 32) {
    float x = p[k];
    s += x * x;
  }
#pragma unroll
  for (int off = 16; off > 0; off >>= 1) s += __shfl_xor(s, off, 32);
  if (lane == 0) dst[row] = s;
}

// ---------------------------------------------------------------- init u,v=0
__global__ void init_uv(float* __restrict__ u, float* __restrict__ v) {
  int j = blockIdx.x * 256 + threadIdx.x;
  u[j] = 0.0f;
  v[j] = 0.0f;
}

// ------------------------------------------- Cf = sqdist(XP,XQ) via WMMA f32
// 128x128 tile per WG (256 threads = 8 waves, 4x2 wave grid, 32x64 per wave)
__global__ __launch_bounds__(256) void gemm_cf(
    const float* __restrict__ XP, const float* __restrict__ XQ,
    const float* __restrict__ nxp, const float* __restrict__ nxq,
    float* __restrict__ Cf, float* __restrict__ pmax) {
  __shared__ float As[128 * 20];
  __shared__ float Bs[128 * 20];
  __shared__ float red[256];
  const int tid = threadIdx.x;
  const int lane = tid & 31;
  const int wave = tid >> 5;
  const int rb = blockIdx.y * 128;
  const int cb = blockIdx.x * 128;
  const int wr = (wave & 3) * 32;
  const int wc = (wave >> 2) * 64;

  v8f acc[2][4];
#pragma unroll
  for (int i = 0; i < 2; i++)
#pragma unroll
    for (int j = 0; j < 4; j++) acc[i][j] = {};

  const int lr = tid >> 2;       // 0..63
  const int cg = (tid & 3) * 4;  // 0,4,8,12
  const int khalf2 = (lane >> 4) * 2;
  const int lm = lane & 15;

  for (int k0 = 0; k0 < kD; k0 += 16) {
    __syncthreads();
#pragma unroll
    for (int rr = 0; rr < 128; rr += 64) {
      v4f va = *(const v4f*)(XP + (size_t)(rb + lr + rr) * kD + k0 + cg);
      *(v4f*)(&As[(lr + rr) * 20 + cg]) = va;
      v4f vb = *(const v4f*)(XQ + (size_t)(cb + lr + rr) * kD + k0 + cg);
      *(v4f*)(&Bs[(lr + rr) * 20 + cg]) = vb;
    }
    __syncthreads();
#pragma unroll
    for (int kk = 0; kk < 16; kk += 4) {
      int koff = kk + khalf2;
      v2f af[2], bf[4];
#pragma unroll
      for (int i = 0; i < 2; i++)
        af[i] = *(const v2f*)(&As[(wr + i * 16 + lm) * 20 + koff]);
#pragma unroll
      for (int j = 0; j < 4; j++)
        bf[j] = *(const v2f*)(&Bs[(wc + j * 16 + lm) * 20 + koff]);
#pragma unroll
      for (int i = 0; i < 2; i++)
#pragma unroll
        for (int j = 0; j < 4; j++)
          acc[i][j] = wmma_f32(af[i], bf[j], acc[i][j]);
    }
  }

  // epilogue: Cf = max(0, |xp|^2 + |xq|^2 - 2*dot), track max
  const int mo = 8 * (lane >> 4);
  float lmax = NEG_INF;
#pragma unroll
  for (int i = 0; i < 2; i++)
#pragma unroll
    for (int j = 0; j < 4; j++)
#pragma unroll
      for (int r = 0; r < 8; r++) {
        int row = rb + wr + i * 16 + r + mo;
        int col = cb + wc + j * 16 + lm;
        float c = nxp[row] + nxq[col] - 2.0f * acc[i][j][r];
        c = fmaxf(c, 0.0f);
        Cf[(size_t)row * kN + col] = c;
        lmax = fmaxf(lmax, c);
      }
  red[tid] = lmax;
  __syncthreads();
  for (int s = 128; s > 0; s >>= 1) {
    if (tid < s) red[tid] = fmaxf(red[tid], red[tid + s]);
    __syncthreads();
  }
  if (tid == 0) pmax[blockIdx.y * gridDim.x + blockIdx.x] = red[0];
}

// ---------------------------------------------------------------- beta = max
__global__ __launch_bounds__(256) void reduce_max(
    const float* __restrict__ pmax, int n, float* __restrict__ beta) {
  __shared__ float red[256];
  float m = NEG_INF;
  for (int i = threadIdx.x; i < n; i += 256) m = fmaxf(m, pmax[i]);
  red[threadIdx.x] = m;
  __syncthreads();
  for (int s = 128; s > 0; s >>= 1) {
    if (threadIdx.x < s)
      red[threadIdx.x] = fmaxf(red[threadIdx.x], red[threadIdx.x + s]);
    __syncthreads();
  }
  if (threadIdx.x == 0) beta[0] = red[0];
}

// ------------------- M = -(Cf + beta * Cl)/REG in place, Cl recomputed (K=16)
// 256 cols (one/thread) x 128 rows per WG; coalesced read-modify-write of M.
__global__ __launch_bounds__(256) void apply_labels(
    const float* __restrict__ YP, const float* __restrict__ YQ,
    const float* __restrict__ nyp, const float* __restrict__ nyq,
    const float* __restrict__ beta_p, float* __restrict__ Mmat) {
  __shared__ float yps[128 * 16];
  __shared__ float nyps[128];
  const int tid = threadIdx.x;
  const int col = blockIdx.x * 256 + tid;
  const int rb = blockIdx.y * 128;

  const v4f* srcy = (const v4f*)(YP + (size_t)rb * kDL);
  ((v4f*)yps)[tid] = srcy[tid];
  ((v4f*)yps)[tid + 256] = srcy[tid + 256];
  if (tid < 128) nyps[tid] = nyp[rb + tid];
  __syncthreads();

  const float beta = beta_p[0];
  const float nq = nyq[col];
  const v4f* yq4 = (const v4f*)(YQ + (size_t)col * kDL);
  v4f q0 = yq4[0], q1 = yq4[1], q2 = yq4[2], q3 = yq4[3];

  for (int r = 0; r < 128; r++) {
    const v4f* yp4 = (const v4f*)(&yps[r * kDL]);
    v4f p0 = yp4[0], p1 = yp4[1], p2 = yp4[2], p3 = yp4[3];
    float dot = p0.x * q0.x + p0.y * q0.y + p0.z * q0.z + p0.w * q0.w;
    dot += p1.x * q1.x + p1.y * q1.y + p1.z * q1.z + p1.w * q1.w;
    dot += p2.x * q2.x + p2.y * q2.y + p2.z * q2.z + p2.w * q2.w;
    dot += p3.x * q3.x + p3.y * q3.y + p3.z * q3.z + p3.w * q3.w;
    float cl = fmaxf(nyps[r] + nq - 2.0f * dot, 0.0f);
    size_t off = (size_t)(rb + r) * kN + col;
    float cf = Mmat[off];
    Mmat[off] = -(cf + beta * cl) * 20.0f;  // 1/REG = 20
  }
}

// ------------------------------------------- u_i = lp - LSE_j(M_ij + v_j)
// one WG per row; b128 loads (8 x 4 contiguous elements per thread, 1024 apart)
__global__ __launch_bounds__(256) void row_lse(const float* __restrict__ Mmat,
                                               const float* __restrict__ v,
                                               float* __restrict__ u,
                                               float lp) {
  __shared__ float sm[256], ss[256];
  const int tid = threadIdx.x;
  const size_t row = blockIdx.x;
  const float* mr = Mmat + row * kN;
  float mx = NEG_INF, s = 0.0f;
#pragma unroll 2
  for (int k = 0; k < 8; k++) {
    int jb = (tid + 256 * k) * 4;
    v4f m4 = *(const v4f*)(mr + jb);
    v4f v4 = *(const v4f*)(v + jb);
    lse_update(m4.x + v4.x, mx, s);
    lse_update(m4.y + v4.y, mx, s);
    lse_update(m4.z + v4.z, mx, s);
    lse_update(m4.w + v4.w, mx, s);
  }
  sm[tid] = mx;
  ss[tid] = s;
  __syncthreads();
  for (int st = 128; st > 0; st >>= 1) {
    if (tid < st) {
      float m1 = sm[tid], s1 = ss[tid];
      lse_merge(sm[tid + st], ss[tid + st], m1, s1);
      sm[tid] = m1;
      ss[tid] = s1;
    }
    __syncthreads();
  }
  if (tid == 0) u[row] = lp - (sm[0] + __logf(ss[0]));
}

// ----- column LSE partials: 4 contiguous cols/thread (b128), 128-row chunks
// grid (8 col-blocks of 1024, 64 row-chunks of 128)
__global__ __launch_bounds__(256) void col_lse_partial(
    const float* __restrict__ Mmat, const float* __restrict__ u,
    float* __restrict__ Pm, float* __restrict__ Ps) {
  __shared__ float us[128];
  const int tid = threadIdx.x;
  const int j0 = blockIdx.x * 1024 + tid * 4;
  const int ib = blockIdx.y * 128;
  if (tid < 128) us[tid] = u[ib + tid];
  __syncthreads();
  float mx0 = NEG_INF, mx1 = NEG_INF, mx2 = NEG_INF, mx3 = NEG_INF;
  float s0 = 0.0f, s1 = 0.0f, s2 = 0.0f, s3 = 0.0f;
  const float* base = Mmat + (size_t)ib * kN + j0;
#pragma unroll 2
  for (int r = 0; r < 128; r++) {
    v4f x = *(const v4f*)(base + (size_t)r * kN);
    float ur = us[r];
    lse_update(x.x + ur, mx0, s0);
    lse_update(x.y + ur, mx1, s1);
    lse_update(x.z + ur, mx2, s2);
    lse_update(x.w + ur, mx3, s3);
  }
  v4f om = {mx0, mx1, mx2, mx3};
  v4f os = {s0, s1, s2, s3};
  *(v4f*)(Pm + (size_t)blockIdx.y * kN + j0) = om;
  *(v4f*)(Ps + (size_t)blockIdx.y * kN + j0) = os;
}

// combine 64 partials per column; 4 cols/thread, grid 8
__global__ __launch_bounds__(256) void col_combine(const float* __restrict__ Pm,
                                                   const float* __restrict__ Ps,
                                                   float* __restrict__ v,
                                                   float lq) {
  const int j0 = (blockIdx.x * 256 + threadIdx.x) * 4;
  float mx0 = NEG_INF, mx1 = NEG_INF, mx2 = NEG_INF, mx3 = NEG_INF;
  float s0 = 0.0f, s1 = 0.0f, s2 = 0.0f, s3 = 0.0f;
#pragma unroll 4
  for (int c = 0; c < 64; c++) {
    v4f m2 = *(const v4f*)(Pm + (size_t)c * kN + j0);
    v4f s2v = *(const v4f*)(Ps + (size_t)c * kN + j0);
    lse_merge(m2.x, s2v.x, mx0, s0);
    lse_merge(m2.y, s2v.y, mx1, s1);
    lse_merge(m2.z, s2v.z, mx2, s2);
    lse_merge(m2.w, s2v.w, mx3, s3);
  }
  v4f res;
  res.x = lq - (mx0 + __logf(s0));
  res.y = lq - (mx1 + __logf(s1));
  res.z = lq - (mx2 + __logf(s2));
  res.w = lq - (mx3 + __logf(s3));
  *(v4f*)(v + j0) = res;
}

// --------------- out = T @ B  with T_ij = exp(u_i + M_ij + v_j + ln n) on the
// fly; 128x64 tile per WG, K-chunks of 32 staged in LDS, WMMA f32 16x16x4.
__global__ __launch_bounds__(256) void final_gemm(
    const float* __restrict__ Mmat, const float* __restrict__ u,
    const float* __restrict__ v, const float* __restrict__ B, int bcols,
    float* __restrict__ out, float logn) {
  __shared__ float Ts[128 * 36];
  __shared__ float Bs[64 * 36];
  const int tid = threadIdx.x;
  const int lane = tid & 31;
  const int wave = tid >> 5;
  const int rb = blockIdx.y * 128;
  const int cb = blockIdx.x * 64;
  const int wr = (wave & 3) * 32;
  const int wc = (wave >> 2) * 32;
  const int lm = lane & 15;
  const int khalf2 = (lane >> 4) * 2;

  v8f acc[2][2];
#pragma unroll
  for (int i = 0; i < 2; i++)
#pragma unroll
    for (int j = 0; j < 2; j++) acc[i][j] = {};

  const int tr = tid >> 3;       // 0..31
  const int c4 = (tid & 7) * 4;  // 0..28
  float uu[4];
#pragma unroll
  for (int p = 0; p < 4; p++) uu[p] = u[rb + tr + p * 32] + logn;

  for (int k0 = 0; k0 < kN; k0 += 32) {
    __syncthreads();
    // stage T tile (128 x 32)
#pragma unroll
    for (int p = 0; p < 4; p++) {
      int row = tr + p * 32;
      v4f mv = *(const v4f*)(Mmat + (size_t)(rb + row) * kN + k0 + c4);
      v4f vv = *(const v4f*)(v + k0 + c4);
      v4f t;
      t.x = __expf(uu[p] + mv.x + vv.x);
      t.y = __expf(uu[p] + mv.y + vv.y);
      t.z = __expf(uu[p] + mv.z + vv.z);
      t.w = __expf(uu[p] + mv.w + vv.w);
      *(v4f*)(&Ts[row * 36 + c4]) = t;
    }
    // stage B tile transposed (64 cols x 32 k), zero-pad past bcols
    {
      int br = tid >> 3;          // k row 0..31
      int cbase = (tid & 7) * 8;  // local col 0..56
#pragma unroll
      for (int q = 0; q < 8; q++) {
        int lc = cbase + q;
        int gc = cb + lc;
        float bv = (gc < bcols) ? B[(size_t)(k0 + br) * bcols + gc] : 0.0f;
        Bs[lc * 36 + br] = bv;
      }
    }
    __syncthreads();
#pragma unroll
    for (int kk = 0; kk < 32; kk += 4) {
      int koff = kk + khalf2;
      v2f af[2], bf[2];
#pragma unroll
      for (int i = 0; i < 2; i++)
        af[i] = *(const v2f*)(&Ts[(wr + i * 16 + lm) * 36 + koff]);
#pragma unroll
      for (int j = 0; j < 2; j++)
        bf[j] = *(const v2f*)(&Bs[(wc + j * 16 + lm) * 36 + koff]);
#pragma unroll
      for (int i = 0; i < 2; i++)
#pragma unroll
        for (int j = 0; j < 2; j++)
          acc[i][j] = wmma_f32(af[i], bf[j], acc[i][j]);
    }
  }

  const int mo = 8 * (lane >> 4);
#pragma unroll
  for (int i = 0; i < 2; i++)
#pragma unroll
    for (int j = 0; j < 2; j++)
#pragma unroll
      for (int r = 0; r < 8; r++) {
        int row = rb + wr + i * 16 + r + mo;
        int col = cb + wc + j * 16 + lm;
        if (col < bcols) out[(size_t)row * bcols + col] = acc[i][j][r];
      }
}

// ---------------------------------------------------------------------------
extern "C" void kernel_launch(void* const* d_in, const int* in_sizes, int n_in,
                              void* d_out, int out_size, void* d_ws,
                              size_t ws_size, hipStream_t stream) {
  const float* XP = (const float*)d_in[0];
  const float* XQ = (const float*)d_in[1];
  const float* YP = (const float*)d_in[2];
  const float* YQ = (const float*)d_in[3];
  float* out = (float*)d_out;

  // workspace layout (floats)
  float* w = (float*)d_ws;
  float* Mw = w;                      // 8192*8192
  float* nxp = Mw + (size_t)kN * kN;  // 8192
  float* nxq = nxp + kN;
  float* nyp = nxq + kN;
  float* nyq = nyp + kN;
  float* u = nyq + kN;
  float* vv = u + kN;
  float* Pm = vv + kN;     // 64*8192
  float* Ps = Pm + 64 * kN;  // 64*8192
  float* pmax = Ps + 64 * kN;  // 4096
  float* beta = pmax + 4096;   // 1

  const float lp = -logf((float)kN);
  const float lq = -logf((float)kN);
  const float logn = logf((float)kN);

  // 0) norms
  row_sumsq<<<kN / 8, 256, 0, stream>>>(XP, nxp, kD);
  row_sumsq<<<kN / 8, 256, 0, stream>>>(XQ, nxq, kD);
  row_sumsq<<<kN / 8, 256, 0, stream>>>(YP, nyp, kDL);
  row_sumsq<<<kN / 8, 256, 0, stream>>>(YQ, nyq, kDL);
  init_uv<<<kN / 256, 256, 0, stream>>>(u, vv);

  // 1) C_features (WMMA f32) + per-WG max
  gemm_cf<<<dim3(kN / 128, kN / 128), 256, 0, stream>>>(XP, XQ, nxp, nxq, Mw,
                                                        pmax);
  // 2) beta
  reduce_max<<<1, 256, 0, stream>>>(pmax, (kN / 128) * (kN / 128), beta);

  // 3) M = -(Cf + beta*Cl)/REG in place (Cl recomputed on the fly)
  apply_labels<<<dim3(kN / 256, kN / 128), 256, 0, stream>>>(YP, YQ, nyp, nyq,
                                                             beta, Mw);

  // 4) log-domain Sinkhorn, 50 iterations (online logsumexp, 1 read of M each)
  for (int it = 0; it < kITER; ++it) {
    row_lse<<<kN, 256, 0, stream>>>(Mw, vv, u, lp);
    col_lse_partial<<<dim3(8, 64), 256, 0, stream>>>(Mw, u, Pm, Ps);
    col_combine<<<8, 256, 0, stream>>>(Pm, Ps, vv, lq);
  }

  // 5) TXP = T @ XQ ; TYP = T @ YQ   (T materialized tile-wise in LDS)
  final_gemm<<<dim3(kD / 64, kN / 128), 256, 0, stream>>>(Mw, u, vv, XQ, kD,
                                                          out, logn);
  final_gemm<<<dim3(1, kN / 128), 256, 0, stream>>>(
      Mw, u, vv, YQ, kDL, out + (size_t)kN * kD, logn);
}